// Decoding_model_10230612099663
// MI455X (gfx1250) — compile-verified
//
#include <hip/hip_runtime.h>
#include <stdint.h>

#define M_ROWS 1024
#define N_COLS 2048
#define B_SZ   16
#define K_MAX  64
#define BIGF   1.0e30f
#define NITER  5

// ---------------------------------------------------------------------------
// CDNA5 async global->LDS staging (ASYNCcnt path, CDNA5 ISA)
// ---------------------------------------------------------------------------
__device__ __forceinline__ void async_g2lds_b128(uint32_t lds_byte_off, const void* gptr) {
    asm volatile("global_load_async_to_lds_b128 %0, %1, off"
                 :: "v"(lds_byte_off), "v"((uint64_t)(uintptr_t)gptr)
                 : "memory");
}
__device__ __forceinline__ void wait_async_zero() {
#if defined(__has_builtin)
#if __has_builtin(__builtin_amdgcn_s_wait_asynccnt)
    __builtin_amdgcn_s_wait_asynccnt(0);
#else
    asm volatile("s_wait_asynccnt 0" ::: "memory");
#endif
#else
    asm volatile("s_wait_asynccnt 0" ::: "memory");
#endif
}

// ---------------------------------------------------------------------------
// K1: CSR extraction, one wave32 per check row, ballot compaction
// ---------------------------------------------------------------------------
__global__ void __launch_bounds__(256) k_build_csr(const int* __restrict__ H,
                                                   int* __restrict__ row_cols,
                                                   int* __restrict__ row_deg) {
    const int lane = threadIdx.x & 31;
    const int wave = threadIdx.x >> 5;
    const int m = blockIdx.x * 8 + wave;
    if (m >= M_ROWS) return;
    const int* hrow = H + (size_t)m * N_COLS;
    int* out = row_cols + m * K_MAX;
    int deg = 0;
    for (int c = 0; c < N_COLS; c += 32) {
        const int n = c + lane;
        const int h = hrow[n];
        const unsigned mask = (unsigned)__ballot(h != 0);
        const int pre = __popc(mask & ((1u << lane) - 1u));
        if (h != 0) {
            const int pos = deg + pre;
            if (pos < K_MAX) out[pos] = n;
        }
        deg += __popc(mask);
    }
    if (lane == 0) row_deg[m] = (deg < K_MAX) ? deg : K_MAX;
}

// ---------------------------------------------------------------------------
// K2: per-column edge counts (coalesced across n)
// ---------------------------------------------------------------------------
__global__ void k_col_count(const int* __restrict__ H, int* __restrict__ col_cnt) {
    const int n = blockIdx.x * blockDim.x + threadIdx.x;
    if (n >= N_COLS) return;
    int c = 0;
    for (int m = 0; m < M_ROWS; ++m) c += (H[(size_t)m * N_COLS + n] != 0);
    col_cnt[n] = c;
}

// ---------------------------------------------------------------------------
// K3: exclusive scan of 2048 counts: 256 threads x 8-chunk serial + LDS
// Hillis-Steele over the 256 partials. Single workgroup, deterministic.
// ---------------------------------------------------------------------------
__global__ void __launch_bounds__(256) k_col_scan(const int* __restrict__ col_cnt,
                                                  int* __restrict__ col_ptr) {
    __shared__ int part[256];
    const int t = threadIdx.x;
    const int base = t * 8;
    int local[8];
    int s = 0;
#pragma unroll
    for (int i = 0; i < 8; ++i) { local[i] = s; s += col_cnt[base + i]; }
    part[t] = s;
    __syncthreads();
#pragma unroll
    for (int off = 1; off < 256; off <<= 1) {
        const int u = (t >= off) ? part[t - off] : 0;
        __syncthreads();
        part[t] += u;
        __syncthreads();
    }
    const int excl = (t == 0) ? 0 : part[t - 1];
#pragma unroll
    for (int i = 0; i < 8; ++i) col_ptr[base + i] = excl + local[i];
    if (t == 255) col_ptr[N_COLS] = excl + s;
}

// ---------------------------------------------------------------------------
// K4: CSC fill with CSR-edge ids (ascending-row order -> deterministic sums)
// ---------------------------------------------------------------------------
__global__ void k_fill_csc(const int* __restrict__ H, const int* __restrict__ row_cols,
                           const int* __restrict__ row_deg, const int* __restrict__ col_ptr,
                           int* __restrict__ col_edges) {
    const int n = blockIdx.x * blockDim.x + threadIdx.x;
    if (n >= N_COLS) return;
    int idx = col_ptr[n];
    for (int m = 0; m < M_ROWS; ++m) {
        if (H[(size_t)m * N_COLS + n] != 0) {
            const int* rc = row_cols + m * K_MAX;
            int lo = 0, hi = row_deg[m] - 1, k = -1;
            while (lo <= hi) {                 // row lists are sorted ascending
                const int mid = (lo + hi) >> 1;
                const int v = rc[mid];
                if (v == n) { k = mid; break; }
                if (v < n) lo = mid + 1; else hi = mid - 1;
            }
            col_edges[idx++] = (k >= 0) ? (m * K_MAX + k) : -1; // -1: overflow sentinel
        }
    }
}

// ---------------------------------------------------------------------------
// K5: check-node update. soft[b,:] (8 KB) staged to LDS with async b128
// copies; one thread per check row: sign product + min/2nd-min -> extrinsic cv
// ---------------------------------------------------------------------------
__global__ void __launch_bounds__(256) k_check_node(const float* __restrict__ soft_cur,
                                                    const int* __restrict__ row_cols,
                                                    const int* __restrict__ row_deg,
                                                    const float* __restrict__ wptr,
                                                    float* __restrict__ cv) {
    __shared__ __align__(16) float s_soft[N_COLS];
    const int t = threadIdx.x;
    const int b = blockIdx.y;

    {
        const float* src = soft_cur + (size_t)b * N_COLS + t * 8;
        const uint32_t lds = (uint32_t)(uintptr_t)(&s_soft[t * 8]); // low 32 bits = LDS offset
        async_g2lds_b128(lds, src);
        async_g2lds_b128(lds + 16u, src + 4);
        wait_async_zero();
    }
    __syncthreads();

    const int m = blockIdx.x * 256 + t;
    const int deg = row_deg[m];
    const int* rc = row_cols + m * K_MAX;
    const float norm = log1pf(expf(wptr[0]));   // softplus(w)

    // pass 1: sign product (0 if any exact zero), min1 <= min2 of clipped |v|
    float min1 = 2.0f * BIGF, min2 = 2.0f * BIGF;
    float sgn = 1.0f;
    for (int k = 0; k < deg; ++k) {
        const float v = s_soft[rc[k]];
        sgn *= (v > 0.0f) ? 1.0f : ((v < 0.0f) ? -1.0f : 0.0f);
        const float a = fminf(fabsf(v), BIGF);
        if (a < min1) { min2 = min1; min1 = a; }
        else if (a < min2) { min2 = a; }
    }
    // pass 2: extrinsic message per edge
    float* cvb = cv + ((size_t)b * M_ROWS + (size_t)m) * K_MAX;
    for (int k = 0; k < deg; ++k) {
        const float v = s_soft[rc[k]];
        const float s = (v > 0.0f) ? 1.0f : ((v < 0.0f) ? -1.0f : 0.0f);
        const float a = fminf(fabsf(v), BIGF);
        const float mag = (a > min1) ? min1 : min2;
        cvb[k] = norm * mag * sgn * s;
    }
}

// ---------------------------------------------------------------------------
// K6: variable-node update. One wave32 per (b, n): lanes take a 32-stride
// slice of the CSC list, then a fixed 5-step shfl_xor butterfly combines.
// Handles the degree-1024 columns (H[:, :2] == 1) without a serial tail;
// fixed reduction shape -> bit-identical across graph replays.
// ---------------------------------------------------------------------------
__global__ void __launch_bounds__(256) k_var_node(const float* __restrict__ soft_in,
                                                  const float* __restrict__ cv,
                                                  const int* __restrict__ col_ptr,
                                                  const int* __restrict__ col_edges,
                                                  float* __restrict__ dst) {
    const int lane = threadIdx.x & 31;
    const int wid  = blockIdx.x * 8 + (threadIdx.x >> 5);  // one wave per (b, n)
    const int b = wid >> 11;                // wid / N_COLS
    const int n = wid & (N_COLS - 1);       // wid % N_COLS
    const float* cvb = cv + (size_t)b * (M_ROWS * K_MAX);
    const int e0 = col_ptr[n], e1 = col_ptr[n + 1];
    float acc = 0.0f;
    for (int j = e0 + lane; j < e1; j += 32) {
        const int e = col_edges[j];
        if (e >= 0) acc += cvb[e];
    }
#pragma unroll
    for (int off = 16; off >= 1; off >>= 1)
        acc += __shfl_xor(acc, off, 32);
    if (lane == 0) {
        const size_t idx = (size_t)b * N_COLS + n;
        dst[idx] = soft_in[idx] + acc;
    }
}

// ---------------------------------------------------------------------------
// Launch
// ---------------------------------------------------------------------------
extern "C" void kernel_launch(void* const* d_in, const int* in_sizes, int n_in,
                              void* d_out, int out_size, void* d_ws, size_t ws_size,
                              hipStream_t stream) {
    const float* soft_input = (const float*)d_in[0];
    const int*   H          = (const int*)d_in[1];
    // d_in[2] = labels (unused by the forward reference)
    const float* w          = (const float*)d_in[3];
    float* out = (float*)d_out;

    char* ws = (char*)d_ws;
    size_t off = 0;
    auto alloc = [&](size_t bytes) -> void* {
        void* p = (void*)(ws + off);
        off += (bytes + 255) & ~(size_t)255;
        return p;
    };
    int*   row_cols  = (int*)alloc((size_t)M_ROWS * K_MAX * sizeof(int));   // 256 KB
    int*   row_deg   = (int*)alloc((size_t)M_ROWS * sizeof(int));           //   4 KB
    int*   col_cnt   = (int*)alloc((size_t)N_COLS * sizeof(int));           //   8 KB
    int*   col_ptr   = (int*)alloc((size_t)(N_COLS + 1) * sizeof(int));     //   8 KB
    int*   col_edges = (int*)alloc((size_t)M_ROWS * K_MAX * sizeof(int));   // 256 KB (bound)
    float* cvbuf     = (float*)alloc((size_t)B_SZ * M_ROWS * K_MAX * sizeof(float)); // 4 MB
    float* bufA      = (float*)alloc((size_t)B_SZ * N_COLS * sizeof(float));         // 128 KB
    float* bufB      = (float*)alloc((size_t)B_SZ * N_COLS * sizeof(float));         // 128 KB
    (void)ws_size; (void)in_sizes; (void)n_in; (void)out_size;

    // Sparsity extraction (recomputed every call: deterministic, no cached state)
    k_build_csr<<<M_ROWS / 8,   256, 0, stream>>>(H, row_cols, row_deg);
    k_col_count<<<N_COLS / 256, 256, 0, stream>>>(H, col_cnt);
    k_col_scan <<<1,            256, 0, stream>>>(col_cnt, col_ptr);
    k_fill_csc <<<N_COLS / 256, 256, 0, stream>>>(H, row_cols, row_deg, col_ptr, col_edges);

    // 5 flooding iterations; last var-node pass writes d_out directly
    const float* src = soft_input;
    float* pp[2] = { bufA, bufB };
    for (int it = 0; it < NITER; ++it) {
        k_check_node<<<dim3(M_ROWS / 256, B_SZ), 256, 0, stream>>>(src, row_cols, row_deg, w, cvbuf);
        float* dst = (it == NITER - 1) ? out : pp[it & 1];
        k_var_node<<<(B_SZ * N_COLS) / 8 / 256 * 8, 256, 0, stream>>>(soft_input, cvbuf, col_ptr, col_edges, dst);
        // grid above = B*N/8/256*8 = 128 blocks (8 waves each, 1024 waves)
        src = dst;
    }
}